// E63NonlinearDelta_4887672783122
// MI455X (gfx1250) — compile-verified
//
#include <hip/hip_runtime.h>

// Problem constants (reference: D=1024, T=1024, B=8, fp32).
constexpr int kD = 1024;
constexpr int kT = 1024;
constexpr int kB = 8;
constexpr int kRows = kT * kB;        // 8192 flattened [T,B] rows of x
constexpr int kNT = kD / 16;          // 64 column tiles of 16
constexpr int kRowTiles = kRows / 16; // 512 row tiles
constexpr int kMT = 4;                // row tiles per wave in precompute (64-row strip)

// Scan: 16 blocks x 512 threads. Block owns 64 output columns (4 col-tiles),
// 16 waves = 4 col-tiles x 4 K-splits. Fewer blocks => cheaper grid barrier.
constexpr int kScanBlocks = 16;
constexpr int kScanThreads = 512;
constexpr int kColsPerBlock = kD / kScanBlocks;  // 64

typedef float    v2f  __attribute__((ext_vector_type(2)));
typedef float    v8f  __attribute__((ext_vector_type(8)));
typedef unsigned u32x4 __attribute__((ext_vector_type(4)));
typedef int      i32x4 __attribute__((ext_vector_type(4)));
typedef int      i32x8 __attribute__((ext_vector_type(8)));

// D = A(16x4,f32) * B(4x16,f32) + C(16x16,f32); fp32 WMMA keeps fp32 precision.
__device__ __forceinline__ v8f wmma_f32_16x16x4(v2f a, v2f b, v8f c) {
  return __builtin_amdgcn_wmma_f32_16x16x4_f32(
      false, a, false, b, (short)0, c, false, false);
}

// ---------------------------------------------------------------------------
// TDM: DMA a strided 2D f32 tile (rows x cols, row stride in elements) from
// global memory into LDS. D# built per CDNA5 ISA ch.8 (group0 + group1, 2D).
// This toolchain uses the 6-arg builtin: (g0, g1, g2, g3, g_extra, cpol).
// ---------------------------------------------------------------------------
__device__ __forceinline__ void tdm_load_2d_f32(const float* gptr, unsigned lds_off,
                                                int rows, int cols, int row_stride) {
  const unsigned long long ga = (unsigned long long)(uintptr_t)gptr;
  u32x4 g0;
  g0[0] = 1u;                                             // count=1, user D#
  g0[1] = lds_off;                                        // lds_addr (bytes)
  g0[2] = (unsigned)(ga & 0xFFFFFFFFull);                 // global_addr[31:0]
  g0[3] = (unsigned)((ga >> 32) & 0x1FFFFFFull)           // global_addr[56:32]
          | (2u << 30);                                   // type = 2 ("image")
  i32x8 g1;
  g1[0] = (int)(2u << 16);                                // data_size=4B, wg_mask=0
  g1[1] = (int)((unsigned)(cols & 0xFFFF) << 16);         // tensor_dim0[15:0]
  g1[2] = (int)(((unsigned)cols >> 16) & 0xFFFFu)         // tensor_dim0[31:16]
          | (int)((unsigned)(rows & 0xFFFF) << 16);       // tensor_dim1[15:0]
  g1[3] = (int)((unsigned)(cols & 0xFFFF) << 16);         // dim1 hi=0, tile_dim0=cols
  g1[4] = rows;                                           // tile_dim1=rows, tile_dim2=0
  g1[5] = row_stride;                                     // tensor_dim0_stride[31:0]
  g1[6] = 0;                                              // stride hi + dim1_stride
  g1[7] = 0;
  i32x4 gz4 = {0, 0, 0, 0};                               // 2D: groups 2/3 unused
  i32x8 gz8 = {0, 0, 0, 0, 0, 0, 0, 0};
  __builtin_amdgcn_tensor_load_to_lds(g0, g1, gz4, gz4, gz8, 0);
}

// ---------------------------------------------------------------------------
// Init: seed h[0] = h0 in d_out's h region, zero grid-barrier counters.
// ---------------------------------------------------------------------------
__global__ void init_kernel(const float* __restrict__ h0,
                            float* __restrict__ h,
                            unsigned* __restrict__ bar) {
  int i = blockIdx.x * blockDim.x + threadIdx.x;
  if (i < kB * kD) h[i] = h0[i];
  if (i < 2) bar[i] = 0u;
}

// ---------------------------------------------------------------------------
// Precompute: alpha = sigmoid(x @ W_alpha^T + b_alpha), wx = x @ W_x^T + b.
// One wave owns a 64x16 strip (4 row-tiles) for BOTH weight matrices:
// the two B loads are amortized over 4 A tiles (2x arithmetic intensity vs
// one tile/wave), 8 independent WMMA accumulator chains hide latency.
// ---------------------------------------------------------------------------
__global__ void __launch_bounds__(256)
precompute_kernel(const float* __restrict__ x,
                  const float* __restrict__ Wa, const float* __restrict__ ba,
                  const float* __restrict__ Wx, const float* __restrict__ bx,
                  float* __restrict__ alpha_all, float* __restrict__ wx_all) {
  const int lane = threadIdx.x & 31;
  const int wave = threadIdx.x >> 5;
  const int gw   = blockIdx.x * 8 + wave;        // 0..8191 (strip, coltile)
  const int sm   = gw / kNT;                     // strip 0..127 (64 rows each)
  const int tn   = gw % kNT;                     // col tile 0..63
  const int row0 = sm * (16 * kMT);
  const int col0 = tn * 16;
  const int l15  = lane & 15;
  const int half = lane >> 4;                    // K pair {0,1} vs {2,3}

  const float* xr[kMT];
#pragma unroll
  for (int ti = 0; ti < kMT; ++ti)
    xr[ti] = x + (size_t)(row0 + ti * 16 + l15) * kD;   // A row m = l15
  const float* warow = Wa + (size_t)(col0 + l15) * kD;  // B col n = l15
  const float* wxrow = Wx + (size_t)(col0 + l15) * kD;

  v8f ca[kMT], cx[kMT];
#pragma unroll
  for (int ti = 0; ti < kMT; ++ti) { ca[ti] = (v8f)0.0f; cx[ti] = (v8f)0.0f; }

  for (int k0 = 0; k0 < kD; k0 += 4) {
    const int k = k0 + 2 * half;
    v2f bA, bX;
    bA.x = warow[k]; bA.y = warow[k + 1];
    bX.x = wxrow[k]; bX.y = wxrow[k + 1];
#pragma unroll
    for (int ti = 0; ti < kMT; ++ti) {
      v2f a;
      a.x = xr[ti][k]; a.y = xr[ti][k + 1];
      ca[ti] = wmma_f32_16x16x4(a, bA, ca[ti]);
      cx[ti] = wmma_f32_16x16x4(a, bX, cx[ti]);
    }
  }

  // C layout: lane holds col n=l15, rows i + 8*half for VGPR i.
  const int e = col0 + l15;
  const float biasA = ba[e];
  const float biasX = bx[e];
#pragma unroll
  for (int ti = 0; ti < kMT; ++ti) {
#pragma unroll
    for (int i = 0; i < 8; ++i) {
      const int m = ti * 16 + i + 8 * half;
      const size_t idx = (size_t)(row0 + m) * kD + e;
      const float za = ca[ti][i] + biasA;
      alpha_all[idx] = 1.0f / (1.0f + __expf(-za));
      wx_all[idx]    = cx[ti][i] + biasX;
    }
  }
}

// ---------------------------------------------------------------------------
// Device-wide sense barrier. bar[0]=arrive count, bar[1]=generation.
// ---------------------------------------------------------------------------
__device__ __forceinline__ void grid_barrier(unsigned* bar, unsigned nblocks) {
  __syncthreads();
  if (threadIdx.x == 0) {
    volatile unsigned* gen = bar + 1;
    const unsigned g = *gen;
    __threadfence();
    if (atomicAdd(bar, 1u) == nblocks - 1u) {
      *bar = 0u;
      __threadfence();
      atomicAdd(bar + 1, 1u);      // release
    } else {
      while (*gen == g) __builtin_amdgcn_s_sleep(2);
      __threadfence();             // acquire
    }
  }
  __syncthreads();
}

// ---------------------------------------------------------------------------
// Persistent scan. Block owns columns [64*blk, 64*blk+64): 4 col-tiles x
// 4 K-split waves. Per step: wave0 launches two TDM DMAs (alpha[t], wx[t]
// 8x64 strided tiles -> LDS) which overlap the WMMA K-reduction; partial C
// tiles reduce through LDS; 512 threads apply gate/nonlinearity reading
// alpha/wx from LDS; grid barrier separates steps.
// ---------------------------------------------------------------------------
__global__ void __launch_bounds__(kScanThreads)
scan_kernel(const float* __restrict__ alpha_all,
            const float* __restrict__ wx_all,
            const float* __restrict__ Wh,
            float* __restrict__ outs,
            float* __restrict__ h,          // [T+1, B, D] region of d_out
            unsigned* __restrict__ bar) {
  const int lane = threadIdx.x & 31;
  const int wave = threadIdx.x >> 5;        // 0..15
  const int l15  = lane & 15;
  const int half = lane >> 4;
  const int ct   = wave >> 2;               // col-tile within block, 0..3
  const int kw   = wave & 3;                // K split, 0..3
  const int colb = blockIdx.x * kColsPerBlock;     // block column base
  const int col0 = colb + ct * 16;                 // this wave's col tile

  __shared__ float part[4][4][16][16];      // [ct][kw][m][n]  16 KB
  __shared__ float alphaT[kB * kColsPerBlock];     // 2 KB
  __shared__ float wxT[kB * kColsPerBlock];        // 2 KB

  const float* whrow = Wh + (size_t)(col0 + l15) * kD;  // B col n = l15
  const int kbase = kw * (kD / 4);                      // 256 K per wave
  const bool mvalid = (l15 < kB);                       // A rows 8..15 pad

  for (int t = 0; t < kT; ++t) {
    const float* hprev = h + (size_t)t * (kB * kD);

    // Kick off async tensor DMAs for this step's alpha/wx block tiles.
    if (wave == 0) {
      const size_t base = (size_t)t * kB * kD + colb;
      tdm_load_2d_f32(alpha_all + base,
                      (unsigned)(uintptr_t)&alphaT[0], kB, kColsPerBlock, kD);
      tdm_load_2d_f32(wx_all + base,
                      (unsigned)(uintptr_t)&wxT[0], kB, kColsPerBlock, kD);
    }

    // K-reduction: 64 WMMAs per wave, 4 independent accumulator chains.
    const float* arow = hprev + (size_t)l15 * kD;       // A row m = l15
    v8f c0 = {}, c1 = {}, c2 = {}, c3 = {};
    for (int k0 = kbase; k0 < kbase + kD / 4; k0 += 16) {
      const int kk = k0 + 2 * half;
      v2f a0, a1, a2, a3, b0, b1, b2, b3;
      a0.x = mvalid ? arow[kk + 0]  : 0.0f;  a0.y = mvalid ? arow[kk + 1]  : 0.0f;
      a1.x = mvalid ? arow[kk + 4]  : 0.0f;  a1.y = mvalid ? arow[kk + 5]  : 0.0f;
      a2.x = mvalid ? arow[kk + 8]  : 0.0f;  a2.y = mvalid ? arow[kk + 9]  : 0.0f;
      a3.x = mvalid ? arow[kk + 12] : 0.0f;  a3.y = mvalid ? arow[kk + 13] : 0.0f;
      b0.x = whrow[kk + 0];   b0.y = whrow[kk + 1];
      b1.x = whrow[kk + 4];   b1.y = whrow[kk + 5];
      b2.x = whrow[kk + 8];   b2.y = whrow[kk + 9];
      b3.x = whrow[kk + 12];  b3.y = whrow[kk + 13];
      c0 = wmma_f32_16x16x4(a0, b0, c0);
      c1 = wmma_f32_16x16x4(a1, b1, c1);
      c2 = wmma_f32_16x16x4(a2, b2, c2);
      c3 = wmma_f32_16x16x4(a3, b3, c3);
    }
    v8f c = (c0 + c1) + (c2 + c3);
#pragma unroll
    for (int i = 0; i < 8; ++i) part[ct][kw][i + 8 * half][l15] = c[i];

    if (wave == 0) __builtin_amdgcn_s_wait_tensorcnt(0);  // DMA done (wave0)
    __syncthreads();                                      // publish LDS to all

    // Epilogue: 512 threads -> (b=m in 0..7, c in 0..63), one element each.
    {
      const int m = threadIdx.x >> 6;          // batch index 0..7
      const int cc = threadIdx.x & 63;         // column within block
      const int e = colb + cc;
      const int pc = cc >> 4, pn = cc & 15;
      const float acc = part[pc][0][m][pn] + part[pc][1][m][pn] +
                        part[pc][2][m][pn] + part[pc][3][m][pn];
      const float wx = wxT[m * kColsPerBlock + cc];
      const float al = alphaT[m * kColsPerBlock + cc];
      const float hp = hprev[m * kD + e];
      const float v  = tanhf(acc + wx);
      const float hn = al * hp + (1.0f - al) * v;
      const float sg = 1.0f / (1.0f + __expf(-hn));
      const size_t tbe = ((size_t)t * kB + m) * kD + e;
      outs[tbe] = hn * hn * sg;                             // h * silu(h)
      h[((size_t)(t + 1) * kB + m) * kD + e] = hn;
    }
    __threadfence();
    grid_barrier(bar, kScanBlocks);
  }
}

// ---------------------------------------------------------------------------
// Launch. Inputs: x, h0, W_alpha, b_alpha, W_h, W_x, b.
// d_out = outs [T,B,D] ++ h [T+1,B,D]. d_ws = alpha ++ wx ++ barrier.
// ---------------------------------------------------------------------------
extern "C" void kernel_launch(void* const* d_in, const int* in_sizes, int n_in,
                              void* d_out, int out_size, void* d_ws, size_t ws_size,
                              hipStream_t stream) {
  (void)in_sizes; (void)n_in; (void)out_size; (void)ws_size;

  const float* x  = (const float*)d_in[0];
  const float* h0 = (const float*)d_in[1];
  const float* Wa = (const float*)d_in[2];
  const float* ba = (const float*)d_in[3];
  const float* Wh = (const float*)d_in[4];
  const float* Wx = (const float*)d_in[5];
  const float* bx = (const float*)d_in[6];

  float* outs = (float*)d_out;
  float* h    = outs + (size_t)kT * kB * kD;        // h[0..T]

  float* alpha = (float*)d_ws;
  float* wx    = alpha + (size_t)kT * kB * kD;
  unsigned* bar = (unsigned*)(wx + (size_t)kT * kB * kD);

  init_kernel<<<(kB * kD + 255) / 256, 256, 0, stream>>>(h0, h, bar);

  // 128 strips x 64 col-tiles = 8192 waves, 8 waves per 256-thread block.
  precompute_kernel<<<(kRowTiles / kMT) * kNT / 8, 256, 0, stream>>>(
      x, Wa, ba, Wx, bx, alpha, wx);

  scan_kernel<<<kScanBlocks, kScanThreads, 0, stream>>>(
      alpha, wx, Wh, outs, h, bar);
}